// SinkhornDistance_29411936043718
// MI455X (gfx1250) — compile-verified
//
#include <hip/hip_runtime.h>
#include <math.h>

typedef __attribute__((ext_vector_type(2))) float v2f;
typedef __attribute__((ext_vector_type(8))) float v8f;

#define EPS      0.1f
#define INV_EPS  10.0f
#define SINK_ITERS 50
#define THRESH   0.1f

constexpr int NB = 4;      // batches
constexpr int P  = 1024;   // points per side
constexpr int D  = 64;     // feature dim
constexpr int PP = P * P;  // 1048576

struct Ws {
    float u[NB * P];
    float v[NB * P];
    float x2[NB * P];
    float y2[NB * P];
    float du[NB * P];        // per-row |u_new - u| for deterministic err reduce
    float part[NB * 512];    // per-block partial costs (k_pi), fixed-order reduced
    int   done;
};

// ---------------------------------------------------------------- init
__global__ void k_init(Ws* ws) {
    int i = blockIdx.x * blockDim.x + threadIdx.x;
    if (i < NB * P) { ws->u[i] = 0.0f; ws->v[i] = 0.0f; }
    if (i == 0) ws->done = 0;
}

// ---------------------------------------------------------------- row norms
__global__ void k_rownorm(const float* __restrict__ x, const float* __restrict__ y, Ws* ws) {
    int i = blockIdx.x * blockDim.x + threadIdx.x;   // 0 .. 2*NB*P-1
    if (i >= 2 * NB * P) return;
    bool isx = (i < NB * P);
    int r = isx ? i : (i - NB * P);
    const float* src = (isx ? x : y) + (size_t)r * D;
    float s = 0.0f;
    #pragma unroll 8
    for (int d = 0; d < D; ++d) s += src[d] * src[d];
    if (isx) ws->x2[r] = s; else ws->y2[r] = s;
}

// ---------------------------------------------------------------- C = |x-y|^2 via f32 WMMA
// one wave per 16x16 tile; NB*64*64 = 16384 tiles
__global__ void k_cost(const float* __restrict__ x, const float* __restrict__ y,
                       const Ws* __restrict__ ws, float* __restrict__ Cout) {
    int wave = (blockIdx.x * blockDim.x + threadIdx.x) >> 5;
    int lane = threadIdx.x & 31;
    const int tpb = P / 16;                 // 64 tiles per dim
    int n  = wave / (tpb * tpb);
    int t  = wave % (tpb * tpb);
    int p0 = (t / tpb) * 16;
    int q0 = (t % tpb) * 16;

    const float* xb = x + (size_t)n * P * D;
    const float* yb = y + (size_t)n * P * D;

    int row   = lane & 15;                  // A's M, B's N, D's N
    int khalf = (lane >> 4) * 2;            // K sub-offset per VGPR-layout

    const float* xrow = xb + (size_t)(p0 + row) * D + khalf;
    const float* yrow = yb + (size_t)(q0 + row) * D + khalf;

    v8f acc = {};
    #pragma unroll
    for (int k = 0; k < D; k += 4) {
        v2f a = *(const v2f*)(xrow + k);
        v2f b = *(const v2f*)(yrow + k);
        // D = A(16x4 f32) * B(4x16 f32) + C, exact fp32 matrix op
        acc = __builtin_amdgcn_wmma_f32_16x16x4_f32(
                  false, a, false, b, (short)0, acc, false, false);
    }

    float y2c = ws->y2[n * P + q0 + row];
    float* Cb = Cout + (size_t)n * PP;
    #pragma unroll
    for (int r = 0; r < 8; ++r) {
        int m = r + 8 * (lane >> 4);        // C/D VGPR layout
        float val = ws->x2[n * P + p0 + m] + y2c - 2.0f * acc[r];
        val = fmaxf(val, 0.0f);
        Cb[(size_t)(p0 + m) * P + (q0 + row)] = val;
    }
}

// ---------------------------------------------------------------- u update (row LSE)
// one wave per row; u_new = eps*(log_mu - LSE_q((v[q]-C[p,q])/eps))
__global__ void k_u(const float* __restrict__ Cmat, Ws* ws) {
    int wid  = (blockIdx.x * blockDim.x + threadIdx.x) >> 5;  // 0..NB*P-1
    int lane = threadIdx.x & 31;
    int n = wid >> 10;
    int p = wid & (P - 1);
    const float* Crow = Cmat + (size_t)n * PP + (size_t)p * P;
    const float* vv   = ws->v + n * P;

    float m = -INFINITY, s = 0.0f;
    for (int q = lane; q < P; q += 32) {            // coalesced across lanes
        float val = (vv[q] - Crow[q]) * INV_EPS;
        float nm  = fmaxf(m, val);
        s = s * __expf(m - nm) + __expf(val - nm);
        m = nm;
    }
    #pragma unroll
    for (int off = 16; off > 0; off >>= 1) {        // fixed-tree merge (deterministic)
        float m2 = __shfl_xor(m, off);
        float s2 = __shfl_xor(s, off);
        float nm = fmaxf(m, m2);
        s = s * __expf(m - nm) + s2 * __expf(m2 - nm);
        m = nm;
    }
    if (lane == 0) {
        float lse = m + __logf(s);
        const float log_mu = __logf(1.0f / P + 1e-8f);
        float un = EPS * (log_mu - lse);
        ws->du[wid] = fabsf(un - ws->u[wid]);
        if (!ws->done) ws->u[wid] = un;
    }
}

// ---------------------------------------------------------------- v update (col LSE)
// block(256): 32 columns of one batch, 8 p-slices of 128 rows, LDS merge
__global__ void k_v(const float* __restrict__ Cmat, Ws* ws) {
    int n     = blockIdx.x >> 5;            // 32 blocks per batch
    int q     = (blockIdx.x & 31) * 32 + (threadIdx.x & 31);
    int slice = threadIdx.x >> 5;           // 0..7
    const float* Cb = Cmat + (size_t)n * PP;
    const float* uu = ws->u + n * P;

    float m = -INFINITY, s = 0.0f;
    int p0 = slice * 128;
    for (int p = p0; p < p0 + 128; ++p) {   // lanes hit consecutive q: coalesced
        float val = (uu[p] - Cb[(size_t)p * P + q]) * INV_EPS;
        float nm  = fmaxf(m, val);
        s = s * __expf(m - nm) + __expf(val - nm);
        m = nm;
    }
    __shared__ float sm[8][32], ss[8][32];
    sm[slice][threadIdx.x & 31] = m;
    ss[slice][threadIdx.x & 31] = s;
    __syncthreads();
    if (slice == 0) {
        float M = sm[0][threadIdx.x & 31], S = ss[0][threadIdx.x & 31];
        #pragma unroll
        for (int k = 1; k < 8; ++k) {
            float m2 = sm[k][threadIdx.x & 31], s2 = ss[k][threadIdx.x & 31];
            float nm = fmaxf(M, m2);
            S = S * __expf(M - nm) + s2 * __expf(m2 - nm);
            M = nm;
        }
        float lse = M + __logf(S);
        const float log_nu = __logf(1.0f / P + 1e-8f);
        float vn = EPS * (log_nu - lse);
        if (!ws->done) ws->v[n * P + q] = vn;
    }
}

// ---------------------------------------------------------------- early-stop flag
// single wave, fixed-order reduction -> deterministic across replays
__global__ void k_done(Ws* ws) {
    int lane = threadIdx.x;
    float s = 0.0f;
    for (int i = lane; i < NB * P; i += 32) s += ws->du[i];
    #pragma unroll
    for (int off = 16; off > 0; off >>= 1) s += __shfl_xor(s, off);
    if (lane == 0) {
        if (s * (1.0f / NB) < THRESH) ws->done = 1;
    }
}

// ---------------------------------------------------------------- pi + partial cost
// block(256): 2048 contiguous elements; 512 blocks per batch
__global__ void k_pi(const float* __restrict__ Cmat, Ws* ws, float* __restrict__ pi) {
    int n = blockIdx.x >> 9;
    size_t base = (size_t)(blockIdx.x & 511) * 2048;
    const float* Cb  = Cmat + (size_t)n * PP;
    float* pib       = pi   + (size_t)n * PP;
    const float* uu  = ws->u + n * P;
    const float* vv  = ws->v + n * P;

    float acc = 0.0f;
    #pragma unroll
    for (int k = 0; k < 8; ++k) {
        size_t idx = base + (size_t)k * 256 + threadIdx.x;   // coalesced
        int p = (int)(idx >> 10);
        int q = (int)(idx & (P - 1));
        float c  = Cb[idx];
        float pv = __expf((uu[p] + vv[q] - c) * INV_EPS);
        pib[idx] = pv;
        acc += pv * c;
    }
    #pragma unroll
    for (int off = 16; off > 0; off >>= 1) acc += __shfl_xor(acc, off);
    __shared__ float red[8];
    if ((threadIdx.x & 31) == 0) red[threadIdx.x >> 5] = acc;
    __syncthreads();
    if (threadIdx.x == 0) {
        float t = 0.0f;
        #pragma unroll
        for (int k = 0; k < 8; ++k) t += red[k];
        ws->part[blockIdx.x] = t;            // no float atomics: deterministic
    }
}

// ---------------------------------------------------------------- final cost^(1/2)
// 4 waves, wave n reduces batch n's 512 partials in fixed order
__global__ void k_final(const Ws* __restrict__ ws, float* __restrict__ out) {
    int n    = threadIdx.x >> 5;
    int lane = threadIdx.x & 31;
    float s = 0.0f;
    for (int i = lane; i < 512; i += 32) s += ws->part[n * 512 + i];
    #pragma unroll
    for (int off = 16; off > 0; off >>= 1) s += __shfl_xor(s, off);
    if (lane == 0) out[n] = sqrtf(s);        // P_EXP = 2
}

// ---------------------------------------------------------------- launch
extern "C" void kernel_launch(void* const* d_in, const int* in_sizes, int n_in,
                              void* d_out, int out_size, void* d_ws, size_t ws_size,
                              hipStream_t stream) {
    const float* x = (const float*)d_in[0];
    const float* y = (const float*)d_in[1];
    float* out      = (float*)d_out;
    float* cost_out = out;                          // (4,)
    float* pi_out   = out + NB;                     // (4,1024,1024)
    float* C_out    = out + NB + (size_t)NB * PP;   // (4,1024,1024)
    Ws* ws = (Ws*)d_ws;

    k_init   <<<(NB * P + 255) / 256, 256, 0, stream>>>(ws);
    k_rownorm<<<(2 * NB * P + 255) / 256, 256, 0, stream>>>(x, y, ws);
    // NB*64*64 = 16384 tiles, 8 waves per 256-thread block -> 2048 blocks
    k_cost   <<<2048, 256, 0, stream>>>(x, y, ws, C_out);

    for (int it = 0; it < SINK_ITERS; ++it) {
        k_u   <<<512, 256, 0, stream>>>(C_out, ws);   // 4096 waves, 1 per row
        k_v   <<<128, 256, 0, stream>>>(C_out, ws);   // 32 cols x 8 slices per block
        k_done<<<1, 32, 0, stream>>>(ws);
    }

    k_pi   <<<NB * 512, 256, 0, stream>>>(C_out, ws, pi_out);
    k_final<<<1, NB * 32, 0, stream>>>(ws, cost_out);
}